// ButterworthFilter_2808908612107
// MI455X (gfx1250) — compile-verified
//
#include <hip/hip_runtime.h>
#include <hip/hip_bf16.h>

// ---------------- problem constants (from reference) ----------------
constexpr int T_LEN = 65536;          // samples per row
constexpr int ROWS  = 1024;           // B*C = 16*64 independent rows
constexpr int L     = 1024;           // chunk length
constexpr int NC    = T_LEN / L;      // 64 chunks per row
constexpr int TC    = ROWS * NC;      // 65536 total chunks
constexpr int S     = 8;              // filter state size (order-8 IIR)
constexpr int TILE  = 32;             // time-steps staged per LDS tile (K1)
constexpr int CH    = 256;            // chunks (threads) per block in K1
constexpr int WPAD  = 9;              // padded row stride of W in LDS (K3)

typedef float v2f __attribute__((ext_vector_type(2)));
typedef float v8f __attribute__((ext_vector_type(8)));

__device__ __forceinline__ void load_coeffs(const float* __restrict__ bP,
                                            const float* __restrict__ aP,
                                            float& b0, float* bt, float* at) {
  const float inv = 1.0f / aP[0];
  b0 = bP[0] * inv;
#pragma unroll
  for (int i = 0; i < S; ++i) {
    bt[i] = bP[i + 1] * inv;
    at[i] = aP[i + 1] * inv;
  }
}

// =====================================================================
// K1: per-chunk zero-initial-state filter.
//   - 1 thread per chunk, 256 chunks per block.
//   - x staged through LDS so global traffic is fully coalesced
//     (per-chunk streams are 4KB apart otherwise).
//   - In-place LDS reuse: y overwrites x slot after the read.
//   - Writes y_local to d_out and the chunk's final 8-state to ws.
// =====================================================================
__global__ __launch_bounds__(256) void k1_chunk_filter(
    const float* __restrict__ x, const float* __restrict__ bP,
    const float* __restrict__ aP, float* __restrict__ y,
    float* __restrict__ f_state) {
  __shared__ float buf[CH * (TILE + 1)];  // 256*33*4 = 33 KB
  const int tid = threadIdx.x;
  const size_t chunk0 = (size_t)blockIdx.x * CH;
  const float* xs = x + chunk0 * L;
  float* ys = y + chunk0 * L;

  float b0, bt[S], at[S];
  load_coeffs(bP, aP, b0, bt, at);

  float z[S];
#pragma unroll
  for (int i = 0; i < S; ++i) z[i] = 0.0f;

  constexpr int NT = L / TILE;  // 32 tiles
  for (int tile = 0; tile < NT; ++tile) {
    const size_t tbase = (size_t)tile * TILE;
    // cooperative coalesced load: 256 chunks x TILE floats, float4 granules
#pragma unroll
    for (int i = 0; i < (CH * TILE / 4) / 256; ++i) {  // 8 iters
      const int q = tid + 256 * i;
      const int cl = q >> 3;          // chunk-local  (TILE/4 = 8 float4/chunk)
      const int w = (q & 7) << 2;     // float offset in tile
      const float4 v =
          *reinterpret_cast<const float4*>(xs + (size_t)cl * L + tbase + w);
      float* d = &buf[cl * (TILE + 1) + w];
      d[0] = v.x; d[1] = v.y; d[2] = v.z; d[3] = v.w;
    }
    __syncthreads();

    if (tile + 1 < NT)  // hint the next tile into cache (global_prefetch_b8)
      __builtin_prefetch(xs + (size_t)tid * L + tbase + TILE, 0, 1);

    // sequential transposed direct-form II over this tile (private row)
    float* row = &buf[tid * (TILE + 1)];
#pragma unroll 8
    for (int t = 0; t < TILE; ++t) {
      const float xt = row[t];
      const float yv = fmaf(b0, xt, z[0]);
#pragma unroll
      for (int i = 0; i < S - 1; ++i)
        z[i] = fmaf(xt, bt[i], fmaf(-yv, at[i], z[i + 1]));
      z[S - 1] = fmaf(xt, bt[S - 1], -yv * at[S - 1]);
      row[t] = yv;  // in-place: y replaces x
    }
    __syncthreads();

    // cooperative coalesced store of y_local
#pragma unroll
    for (int i = 0; i < (CH * TILE / 4) / 256; ++i) {
      const int q = tid + 256 * i;
      const int cl = q >> 3;
      const int w = (q & 7) << 2;
      const float* s = &buf[cl * (TILE + 1) + w];
      float4 v; v.x = s[0]; v.y = s[1]; v.z = s[2]; v.w = s[3];
      *reinterpret_cast<float4*>(ys + (size_t)cl * L + tbase + w) = v;
    }
    __syncthreads();
  }

  const size_t g = chunk0 + tid;
#pragma unroll
  for (int i = 0; i < S; ++i) f_state[g * S + i] = z[i];
}

// =====================================================================
// K2a: build W (L x 8 homogeneous output responses, W[t][j] = (A^t e_j)[0])
//      and M = A^L (8x8) by 10 repeated squarings. Tiny; one warp.
// =====================================================================
__global__ void k2a_build(const float* __restrict__ bP,
                          const float* __restrict__ aP,
                          float* __restrict__ Wmat, float* __restrict__ Mmat) {
  float b0, bt[S], at[S];
  load_coeffs(bP, aP, b0, bt, at);
  const int j = threadIdx.x;
  if (j < S) {
    float z[S];
#pragma unroll
    for (int i = 0; i < S; ++i) z[i] = (i == j) ? 1.0f : 0.0f;
    for (int t = 0; t < L; ++t) {
      const float yv = z[0];
      Wmat[t * S + j] = yv;
#pragma unroll
      for (int i = 0; i < S - 1; ++i) z[i] = fmaf(-yv, at[i], z[i + 1]);
      z[S - 1] = -yv * at[S - 1];
    }
  }
  if (j == 0) {
    float P[S * S], Q[S * S];
#pragma unroll
    for (int i = 0; i < S; ++i)
#pragma unroll
      for (int c = 0; c < S; ++c)
        P[i * S + c] = ((c == 0) ? -at[i] : 0.0f) + ((c == i + 1) ? 1.0f : 0.0f);
    for (int k = 0; k < 10; ++k) {  // A^(2^10) = A^1024 = A^L
      for (int i = 0; i < S; ++i)
        for (int c = 0; c < S; ++c) {
          float acc = 0.0f;
#pragma unroll
          for (int m = 0; m < S; ++m) acc = fmaf(P[i * S + m], P[m * S + c], acc);
          Q[i * S + c] = acc;
        }
      for (int e = 0; e < S * S; ++e) P[e] = Q[e];
    }
    for (int e = 0; e < S * S; ++e) Mmat[e] = P[e];
  }
}

// =====================================================================
// K2b: per-row scan of chunk-entry states: z[c] = M z[c-1] + f[c-1]
// =====================================================================
__global__ __launch_bounds__(256) void k2b_scan(
    const float* __restrict__ f_state, const float* __restrict__ Mmat,
    float* __restrict__ z_start) {
  __shared__ float Ms[S * S];
  if (threadIdx.x < S * S) Ms[threadIdx.x] = Mmat[threadIdx.x];
  __syncthreads();
  const int r = blockIdx.x * 256 + threadIdx.x;  // row 0..1023
  float z[S];
#pragma unroll
  for (int i = 0; i < S; ++i) z[i] = 0.0f;
  for (int c = 0; c < NC; ++c) {
    const size_t g = (size_t)r * NC + c;
#pragma unroll
    for (int i = 0; i < S; ++i) z_start[g * S + i] = z[i];
    float zn[S];
#pragma unroll
    for (int i = 0; i < S; ++i) {
      float acc = f_state[g * S + i];
#pragma unroll
      for (int jj = 0; jj < S; ++jj) acc = fmaf(Ms[i * S + jj], z[jj], acc);
      zn[i] = acc;
    }
#pragma unroll
    for (int i = 0; i < S; ++i) z[i] = zn[i];
  }
}

// =====================================================================
// K3: WMMA correction  Y(16t x 16chunks) += W(16x8) x Zstart(8x16)
//   - one wave per 16 chunks; loops 64 time-tiles of 16 rows
//   - two V_WMMA_F32_16X16X4_F32 per tile (K = states 0..3, 4..7)
//   - operand packing follows ISA 32-bit 16x4 A / 16x16 C-D layouts
// =====================================================================
__global__ __launch_bounds__(256) void k3_correct_wmma(
    const float* __restrict__ Wmat, const float* __restrict__ z_start,
    float* __restrict__ y) {
  __shared__ float sW[L * WPAD];  // 1024*9*4 = 36 KB, bank-conflict-free reads
  const int tid = threadIdx.x;
#pragma unroll
  for (int i = 0; i < (L * S) / 256; ++i) {  // 32 iters, coalesced
    const int e = tid + 256 * i;
    sW[(e >> 3) * WPAD + (e & 7)] = Wmat[e];
  }
  __syncthreads();

  const int wave = tid >> 5;
  const int lane = tid & 31;
  const int n = lane & 15;        // N (chunk) and M (time) index = lane%16
  const int hi = lane >> 4;       // lane group selects K pair / M+8 half
  const int kb = hi ? 2 : 0;      // A/B K-pair base per ISA 16x4 layout
  const size_t chbase = ((size_t)blockIdx.x * 8 + wave) * 16;

  // B operand: Z (8 x 16 chunks), constant over the whole t loop
  const float* zrow = z_start + (chbase + n) * S;
  v2f zb1, zb2;
  zb1.x = zrow[kb];     zb1.y = zrow[kb + 1];
  zb2.x = zrow[kb + 4]; zb2.y = zrow[kb + 5];

  float* ybase = y + (chbase + n) * L + (hi ? 8 : 0);

  for (int t0 = 0; t0 < L; t0 += 16) {
    // A operand: W tile (16 time-rows x 8 states), from padded LDS
    const float* wrow = &sW[(t0 + n) * WPAD + kb];
    v2f wa1, wa2;
    wa1.x = wrow[0]; wa1.y = wrow[1];
    wa2.x = wrow[4]; wa2.y = wrow[5];

    // C operand: existing y_local 16x16 tile (8 contiguous floats per lane)
    float* p = ybase + t0;
    const float4 c0 = *reinterpret_cast<const float4*>(p);
    const float4 c1 = *reinterpret_cast<const float4*>(p + 4);
    v8f c = {c0.x, c0.y, c0.z, c0.w, c1.x, c1.y, c1.z, c1.w};

    c = __builtin_amdgcn_wmma_f32_16x16x4_f32(false, wa1, false, zb1,
                                              (short)0, c, false, false);
    c = __builtin_amdgcn_wmma_f32_16x16x4_f32(false, wa2, false, zb2,
                                              (short)0, c, false, false);

    float4 d0, d1;
    d0.x = c[0]; d0.y = c[1]; d0.z = c[2]; d0.w = c[3];
    d1.x = c[4]; d1.y = c[5]; d1.z = c[6]; d1.w = c[7];
    *reinterpret_cast<float4*>(p) = d0;
    *reinterpret_cast<float4*>(p + 4) = d1;
  }
}

// =====================================================================
extern "C" void kernel_launch(void* const* d_in, const int* in_sizes, int n_in,
                              void* d_out, int out_size, void* d_ws,
                              size_t ws_size, hipStream_t stream) {
  const float* x = (const float*)d_in[0];
  const float* b = (const float*)d_in[1];
  const float* a = (const float*)d_in[2];
  float* y = (float*)d_out;

  float* ws = (float*)d_ws;
  float* f_state = ws;                       // TC*8 floats (2 MB)
  float* z_start = ws + (size_t)TC * S;      // TC*8 floats (2 MB)
  float* Wmat = ws + (size_t)2 * TC * S;     // L*8 floats  (32 KB)
  float* Mmat = Wmat + (size_t)L * S;        // 64 floats

  // Pass 1: zero-state chunk filtering + final states
  k1_chunk_filter<<<TC / CH, 256, 0, stream>>>(x, b, a, y, f_state);
  // Build homogeneous responses W and A^L
  k2a_build<<<1, 32, 0, stream>>>(b, a, Wmat, Mmat);
  // Scan chunk-entry states per row
  k2b_scan<<<ROWS / 256, 256, 0, stream>>>(f_state, Mmat, z_start);
  // WMMA rank-8 correction: y += W * z_start
  k3_correct_wmma<<<TC / (16 * 8), 256, 0, stream>>>(Wmat, z_start, y);
}